// GATLayer_70274254897294
// MI455X (gfx1250) — compile-verified
//
#include <hip/hip_runtime.h>

#define ND_N 20000
#define NW_N 30000
#define E_N  500000
#define D_N  256
#define DOUT_N 256

typedef __attribute__((ext_vector_type(2))) float v2f;
typedef __attribute__((ext_vector_type(8))) float v8f;

// ---------- order-preserving float<->uint mapping for atomic max ----------
__device__ __forceinline__ unsigned mapf(float f) {
  unsigned u = __float_as_uint(f);
  return (u & 0x80000000u) ? ~u : (u | 0x80000000u);
}
__device__ __forceinline__ float unmapf(unsigned u) {
  unsigned v = (u & 0x80000000u) ? (u & 0x7fffffffu) : ~u;
  return __uint_as_float(v);
}
#define NEG_INF_MAPPED 0x007fffffu  // mapf(-inf)

// ---------- init ----------
__global__ void fill_u32_kernel(unsigned* __restrict__ p, unsigned v, int n) {
  int i = blockIdx.x * blockDim.x + threadIdx.x;
  if (i < n) p[i] = v;
}

// ---------- per-node dot products against two 256-float weight slices ----------
// one wave per node; lane k-stride 32 for coalescing
__global__ void node_dots_kernel(const float* __restrict__ hid, int n,
                                 const float* __restrict__ wa,
                                 const float* __restrict__ wb,
                                 float* __restrict__ outa,
                                 float* __restrict__ outb) {
  int wave = threadIdx.x >> 5, lane = threadIdx.x & 31;
  int node = blockIdx.x * 8 + wave;
  if (node >= n) return;
  const float* row = hid + (size_t)node * D_N;
  float da = 0.f, db = 0.f;
#pragma unroll
  for (int i = 0; i < 8; ++i) {
    int k = lane + i * 32;
    float x = row[k];
    da += x * wa[k];
    db += x * wb[k];
  }
#pragma unroll
  for (int off = 16; off; off >>= 1) {
    da += __shfl_xor(da, off, 32);
    db += __shfl_xor(db, off, 32);
  }
  if (lane == 0) { outa[node] = da; outb[node] = db; }
}

// ---------- edge pass A: score = leaky_relu(s_src[src]+s_dst[dst]+b), atomic max per dst ----------
__global__ void edge_score_max_kernel(const int* __restrict__ src_idx,
                                      const int* __restrict__ dst_idx,
                                      const float* __restrict__ s_src,
                                      const float* __restrict__ s_dst,
                                      const float* __restrict__ bptr,
                                      float* __restrict__ e_arr,
                                      unsigned* __restrict__ m_arr, int ne) {
  int e = blockIdx.x * blockDim.x + threadIdx.x;
  if (e >= ne) return;
  float a = s_src[src_idx[e]] + s_dst[dst_idx[e]] + bptr[0];
  float v = a > 0.f ? a : 0.01f * a;
  e_arr[e] = v;
  atomicMax(&m_arr[dst_idx[e]], mapf(v));
}

// ---------- edge pass B: exp(e - max[dst]), atomic sum per dst ----------
__global__ void edge_exp_sum_kernel(const int* __restrict__ dst_idx,
                                    const unsigned* __restrict__ m_arr,
                                    float* __restrict__ e_arr,
                                    float* __restrict__ s_arr, int ne) {
  int e = blockIdx.x * blockDim.x + threadIdx.x;
  if (e >= ne) return;
  int d = dst_idx[e];
  float ex = __expf(e_arr[e] - unmapf(m_arr[d]));
  e_arr[e] = ex;
  atomicAdd(&s_arr[d], ex);
}

// ---------- edge pass C: h[dst] += alpha * feat[src]  (one wave per edge) ----------
__global__ void edge_aggregate_kernel(const int* __restrict__ src_idx,
                                      const int* __restrict__ dst_idx,
                                      const float* __restrict__ e_arr,
                                      const float* __restrict__ s_arr,
                                      const float* __restrict__ src_feat,
                                      float* __restrict__ dst_accum, int ne) {
  int wave = threadIdx.x >> 5, lane = threadIdx.x & 31;
  int e = blockIdx.x * 8 + wave;
  if (e >= ne) return;
  int s = src_idx[e], d = dst_idx[e];
  float alpha = e_arr[e] / fmaxf(s_arr[d], 1e-30f);
  const float4* srow = (const float4*)(src_feat + (size_t)s * D_N);
  float* drow = dst_accum + (size_t)d * D_N;
#pragma unroll
  for (int h = 0; h < 2; ++h) {
    float4 v = srow[lane + h * 32];
    int c = (lane + h * 32) * 4;
    atomicAdd(&drow[c + 0], alpha * v.x);
    atomicAdd(&drow[c + 1], alpha * v.y);
    atomicAdd(&drow[c + 2], alpha * v.z);
    atomicAdd(&drow[c + 3], alpha * v.w);
  }
}

// ---------- FC (f32 WMMA 16x16x4) + bias + row softmax ----------
// 128 threads = 4 waves; block handles 16 rows x 256 cols; wave w owns col tiles 4w..4w+3.
__global__ void fc_softmax_kernel(const float* __restrict__ H,
                                  const float* __restrict__ W,
                                  const float* __restrict__ bias,
                                  float* __restrict__ out) {
  __shared__ float tile[16][260];
  int wave = threadIdx.x >> 5;
  int lane = threadIdx.x & 31;
  size_t rowBase = (size_t)blockIdx.x * 16;
  int halfK = lane >> 4;   // 0: K pair {0,1}, 1: K pair {2,3}
  int l16 = lane & 15;

  v8f zero = {0.f, 0.f, 0.f, 0.f, 0.f, 0.f, 0.f, 0.f};
  v8f acc0 = zero, acc1 = zero, acc2 = zero, acc3 = zero;

  const float* hrow = H + (rowBase + l16) * D_N;
  int colBase = wave * 64 + l16;
  for (int k0 = 0; k0 < D_N; k0 += 4) {
    int ka = k0 + halfK * 2;
    v2f a;
    a.x = hrow[ka];
    a.y = hrow[ka + 1];
    const float* w0 = W + (size_t)ka * DOUT_N;
    const float* w1 = w0 + DOUT_N;
    v2f b0; b0.x = w0[colBase +  0]; b0.y = w1[colBase +  0];
    v2f b1; b1.x = w0[colBase + 16]; b1.y = w1[colBase + 16];
    v2f b2; b2.x = w0[colBase + 32]; b2.y = w1[colBase + 32];
    v2f b3; b3.x = w0[colBase + 48]; b3.y = w1[colBase + 48];
    acc0 = __builtin_amdgcn_wmma_f32_16x16x4_f32(false, a, false, b0, (short)0, acc0, false, false);
    acc1 = __builtin_amdgcn_wmma_f32_16x16x4_f32(false, a, false, b1, (short)0, acc1, false, false);
    acc2 = __builtin_amdgcn_wmma_f32_16x16x4_f32(false, a, false, b2, (short)0, acc2, false, false);
    acc3 = __builtin_amdgcn_wmma_f32_16x16x4_f32(false, a, false, b3, (short)0, acc3, false, false);
  }

  // write logits + bias to LDS. C/D layout: lanes 0-15 -> M=r, lanes 16-31 -> M=r+8
  int rowOff = halfK * 8;
  {
    v8f accs[4] = {acc0, acc1, acc2, acc3};
#pragma unroll
    for (int n = 0; n < 4; ++n) {
      int col = colBase + n * 16;
      float bv = bias[col];
#pragma unroll
      for (int r = 0; r < 8; ++r) tile[r + rowOff][col] = accs[n][r] + bv;
    }
  }
  __syncthreads();

  // softmax over each 256-wide row; wave w handles rows w, w+4, w+8, w+12
#pragma unroll
  for (int rr = 0; rr < 4; ++rr) {
    int row = wave + rr * 4;
    float v[8];
    float mx = -3.402823466e38f;
#pragma unroll
    for (int i = 0; i < 8; ++i) {
      v[i] = tile[row][lane + i * 32];
      mx = fmaxf(mx, v[i]);
    }
#pragma unroll
    for (int off = 16; off; off >>= 1) mx = fmaxf(mx, __shfl_xor(mx, off, 32));
    float sum = 0.f;
#pragma unroll
    for (int i = 0; i < 8; ++i) { v[i] = __expf(v[i] - mx); sum += v[i]; }
#pragma unroll
    for (int off = 16; off; off >>= 1) sum += __shfl_xor(sum, off, 32);
    float inv = 1.0f / sum;
    float* orow = out + (rowBase + row) * DOUT_N;
#pragma unroll
    for (int i = 0; i < 8; ++i) orow[lane + i * 32] = v[i] * inv;
  }
}

extern "C" void kernel_launch(void* const* d_in, const int* in_sizes, int n_in,
                              void* d_out, int out_size, void* d_ws, size_t ws_size,
                              hipStream_t stream) {
  const float* doc_hidden  = (const float*)d_in[0];
  const float* word_hidden = (const float*)d_in[1];
  const int*   edge_doc    = (const int*)d_in[2];
  const int*   edge_word   = (const int*)d_in[3];
  const float* d2w_w       = (const float*)d_in[4];  // [512]
  const float* d2w_b       = (const float*)d_in[5];
  const float* w2d_w       = (const float*)d_in[6];  // [512]
  const float* w2d_b       = (const float*)d_in[7];
  const float* fc_w        = (const float*)d_in[8];  // [256,256]
  const float* fc_b        = (const float*)d_in[9];
  float* out = (float*)d_out;

  // ---- workspace carve (floats) ----
  float* ws = (float*)d_ws;
  const size_t NH = (size_t)(ND_N + NW_N) * D_N;      // H: doc rows [0,ND), word rows [ND,ND+NW)
  float*    H        = ws;
  float*    sum_doc  = H + NH;
  float*    sum_word = sum_doc + ND_N;
  unsigned* max_doc  = (unsigned*)(sum_word + NW_N);  // max_doc then max_word contiguous
  unsigned* max_word = max_doc + ND_N;
  float*    sA_doc   = (float*)(max_word + NW_N);     // dot(doc, d2w_w[:256])   (src comp, d2w)
  float*    sB_doc   = sA_doc + ND_N;                 // dot(doc, w2d_w[256:])   (dst comp, w2d)
  float*    sA_word  = sB_doc + ND_N;                 // dot(word, d2w_w[256:])  (dst comp, d2w)
  float*    sB_word  = sA_word + NW_N;                // dot(word, w2d_w[:256])  (src comp, w2d)
  float*    e_d2w    = sB_word + NW_N;
  float*    e_w2d    = e_d2w + E_N;

  // ---- init: zero H + sums (contiguous), set maxes to mapped(-inf) ----
  hipMemsetAsync(ws, 0, (NH + ND_N + NW_N) * sizeof(float), stream);
  {
    int n = ND_N + NW_N;
    fill_u32_kernel<<<(n + 255) / 256, 256, 0, stream>>>(max_doc, NEG_INF_MAPPED, n);
  }

  // ---- per-node score components (one wave per node, 8 nodes per block) ----
  node_dots_kernel<<<ND_N / 8, 256, 0, stream>>>(doc_hidden, ND_N,
                                                 d2w_w, w2d_w + D_N, sA_doc, sB_doc);
  node_dots_kernel<<<NW_N / 8, 256, 0, stream>>>(word_hidden, NW_N,
                                                 d2w_w + D_N, w2d_w, sA_word, sB_word);

  const int egrid = (E_N + 255) / 256;
  // ---- edge pass A: scores + segment max ----
  edge_score_max_kernel<<<egrid, 256, 0, stream>>>(edge_doc, edge_word, sA_doc, sA_word,
                                                   d2w_b, e_d2w, max_word, E_N);
  edge_score_max_kernel<<<egrid, 256, 0, stream>>>(edge_word, edge_doc, sB_word, sB_doc,
                                                   w2d_b, e_w2d, max_doc, E_N);
  // ---- edge pass B: exp + segment sum ----
  edge_exp_sum_kernel<<<egrid, 256, 0, stream>>>(edge_word, max_word, e_d2w, sum_word, E_N);
  edge_exp_sum_kernel<<<egrid, 256, 0, stream>>>(edge_doc, max_doc, e_w2d, sum_doc, E_N);
  // ---- edge pass C: weighted aggregation (one wave per edge) ----
  edge_aggregate_kernel<<<E_N / 8, 256, 0, stream>>>(edge_doc, edge_word, e_d2w, sum_word,
                                                     doc_hidden, H + (size_t)ND_N * D_N, E_N);
  edge_aggregate_kernel<<<E_N / 8, 256, 0, stream>>>(edge_word, edge_doc, e_w2d, sum_doc,
                                                     word_hidden, H, E_N);

  // ---- FC (WMMA f32) + softmax; 50000 rows = 3125 exact 16-row tiles ----
  fc_softmax_kernel<<<(ND_N + NW_N) / 16, 128, 0, stream>>>(H, fc_w, fc_b, out);

  (void)in_sizes; (void)n_in; (void)out_size; (void)ws_size;
}